// SelfSimilar_33741263077459
// MI455X (gfx1250) — compile-verified
//
#include <hip/hip_runtime.h>
#include <hip/hip_bf16.h>

// Problem constants (from reference): K=7, C=32, H=W=192, n=2
constexpr int KS  = 7;
constexpr int PAD = 3;            // KS/2
constexpr int NC  = 32;           // channels
constexpr int NH  = 192;
constexpr int NW  = 192;
constexpr int NQ  = NH * NW * 2;  // 73728 pixel columns (q = (h*W+w)*2 + n)

typedef __attribute__((ext_vector_type(2))) float v2f;
typedef __attribute__((ext_vector_type(8))) float v8f;

// ---------------------------------------------------------------------------
// Kernel A: per-pixel 7x7 depthwise conv + transpose to column-major [q][c].
// Block: 256 threads. Grid: (NW/32, NH).
// Each block handles one h row, a 32-wide w tile, both n, all 32 channels.
//   x  : (C, H, W, 2) float  -- contiguous in (w, n) -> coalesced reads
//   w1 : (49, H, W)   float  -- per-pixel taps, shared across all 64 (n,c)
//   t  : (NQ, 32)     float  -- intermediate, channel-contiguous per column
// ---------------------------------------------------------------------------
__global__ __launch_bounds__(256) void dw_pixelwise_kernel(
    const float* __restrict__ x,
    const float* __restrict__ w1,
    float* __restrict__ t) {
  __shared__ float s_w1[KS * KS * 32];  // [tap][w_local]   6.3 KB
  __shared__ float s_out[64 * 33];      // [wn][c] padded   8.4 KB

  const int tid = threadIdx.x;
  const int h   = blockIdx.y;
  const int wb  = blockIdx.x * 32;

  // Cooperative load of the 49-tap x 32-pixel weight tile (reused by 64 nc).
  for (int idx = tid; idx < KS * KS * 32; idx += 256) {
    const int tap = idx >> 5;
    const int wl  = idx & 31;
    s_w1[idx] = w1[((size_t)tap * NH + h) * NW + wb + wl];
  }
  __syncthreads();

  const int wn = tid & 63;       // local (w,n) index == local q index
  const int wl = wn >> 1;        // w offset 0..31
  const int n  = wn & 1;
  const int w  = wb + wl;
  const int c0 = (tid >> 6) * 8; // 8 channels per thread

  for (int cc = 0; cc < 8; ++cc) {
    const int c = c0 + cc;
    const float* xc = x + (size_t)c * (NH * NW * 2);
    float acc = 0.f;
#pragma unroll
    for (int i = 0; i < KS; ++i) {
      const int hh = h + i - PAD;
      if (hh < 0 || hh >= NH) continue;   // zero padding
      const float* xrow = xc + (size_t)hh * (NW * 2) + n;
#pragma unroll
      for (int j = 0; j < KS; ++j) {
        const int ww = w + j - PAD;
        if (ww < 0 || ww >= NW) continue; // zero padding
        acc = fmaf(xrow[(size_t)ww * 2], s_w1[(i * KS + j) * 32 + wl], acc);
      }
    }
    s_out[wn * 33 + c] = acc;  // stride 33: conflict-free banks
  }
  __syncthreads();

  // Transposed store: lanes run over c -> 128B coalesced per wave.
  const int cl    = tid & 31;
  const int q0    = tid >> 5;             // 0..7
  const int qbase = (h * NW + wb) * 2;
#pragma unroll
  for (int r = 0; r < 8; ++r) {
    const int ql = q0 + r * 8;            // covers 0..63
    t[(size_t)(qbase + ql) * NC + cl] = s_out[ql * 33 + cl];
  }
}

// ---------------------------------------------------------------------------
// Kernel B: 1x1 conv over channels as fp32 WMMA GEMM.
//   out[o, q] = sum_c cw[o, c] * t[q, c] + cb[o]
// Full precision via V_WMMA_F32_16X16X4_F32: K=32 -> 8 chained K=4 steps,
// M=32 -> two 16-row tiles. Each wave32 computes a 32(o) x 16(q) tile.
// fp32 lane layouts (ISA 7.12.2):
//   A 16x4 : lane l -> row M=l%16, regs {K0,K1} = {2*(l>=16), 2*(l>=16)+1}
//   B 4x16 : lane l -> col N=l%16, regs {K0,K1} analogous
//   C/D    : lane l, reg r -> M = r + 8*(l>=16), N = l%16
// Output flat index (c,h,w,2) == o*NQ + q, so D stores straight out.
// Grid: NQ / (8 waves * 16 cols) = 576 blocks of 256 threads.
// ---------------------------------------------------------------------------
__global__ __launch_bounds__(256) void pw_conv_wmma_kernel(
    const float* __restrict__ t,    // [NQ][32]
    const float* __restrict__ cw,   // [32][32] row = out channel
    const float* __restrict__ cb,   // [32]
    float* __restrict__ out) {      // [32][NQ]
  const int lane  = threadIdx.x & 31;
  const int wave  = threadIdx.x >> 5;
  const int half  = lane >> 4;      // 0 = lanes 0-15, 1 = lanes 16-31
  const int l16   = lane & 15;
  const int qbase = (blockIdx.x * 8 + wave) * 16;

  // Accumulators seeded with bias (broadcast along N).
  v8f acc0, acc1;
#pragma unroll
  for (int r = 0; r < 8; ++r) {
    acc0[r] = cb[r + 8 * half];
    acc1[r] = cb[16 + r + 8 * half];
  }

  const float* bcol  = t  + (size_t)(qbase + l16) * NC + 2 * half;
  const float* arow0 = cw + (size_t)l16 * NC + 2 * half;
  const float* arow1 = cw + (size_t)(16 + l16) * NC + 2 * half;

#pragma unroll
  for (int k = 0; k < 8; ++k) {   // K chunks of 4 channels
    v2f a0, a1, b;
    b.x  = bcol[4 * k];      b.y  = bcol[4 * k + 1];
    a0.x = arow0[4 * k];     a0.y = arow0[4 * k + 1];
    a1.x = arow1[4 * k];     a1.y = arow1[4 * k + 1];
    // (neg_a, A, neg_b, B, c_mod, C, reuse_a, reuse_b)
    acc0 = __builtin_amdgcn_wmma_f32_16x16x4_f32(false, a0, false, b,
                                                 (short)0, acc0, false, false);
    acc1 = __builtin_amdgcn_wmma_f32_16x16x4_f32(false, a1, false, b,
                                                 (short)0, acc1, false, false);
  }

  const int q = qbase + l16;
#pragma unroll
  for (int r = 0; r < 8; ++r) {
    out[(size_t)(r + 8 * half) * NQ + q]      = acc0[r];
    out[(size_t)(16 + r + 8 * half) * NQ + q] = acc1[r];
  }
}

// ---------------------------------------------------------------------------
extern "C" void kernel_launch(void* const* d_in, const int* in_sizes, int n_in,
                              void* d_out, int out_size, void* d_ws, size_t ws_size,
                              hipStream_t stream) {
  const float* x  = (const float*)d_in[0];  // (32,192,192,2)
  const float* w1 = (const float*)d_in[1];  // (1,1,49,192,192)
  const float* cw = (const float*)d_in[2];  // (32,32)
  const float* cb = (const float*)d_in[3];  // (32,)
  float* out = (float*)d_out;               // (32,192,192,2) flat = [o][q]
  float* t   = (float*)d_ws;                // intermediate [NQ][32] = 9.4 MB

  dim3 gridA(NW / 32, NH, 1);               // 6 x 192 blocks
  dw_pixelwise_kernel<<<gridA, 256, 0, stream>>>(x, w1, t);

  dim3 gridB(NQ / (8 * 16), 1, 1);          // 576 blocks
  pw_conv_wmma_kernel<<<gridB, 256, 0, stream>>>(t, cw, cb, out);
}